// ManifoldRoPE_58961311040112
// MI455X (gfx1250) — compile-verified
//
#include <hip/hip_runtime.h>
#include <hip/hip_bf16.h>

#define NN 2048
#define HH 16
#define DD 64
#define FF 32
#define PP 16

typedef __attribute__((ext_vector_type(2))) float v2f;
typedef __attribute__((ext_vector_type(8))) float v8f;

// ---------------------------------------------------------------- kernel 1
__global__ __launch_bounds__(256) void mrope_norms(const float* __restrict__ pos,
                                                   float* __restrict__ nrm) {
  int i = blockIdx.x * blockDim.x + threadIdx.x;
  if (i < NN) {
    float s = 0.f;
#pragma unroll
    for (int p = 0; p < PP; ++p) {
      float v = pos[(size_t)i * PP + p];
      s += v * v;
    }
    nrm[i] = s;
  }
}

// ---------------------------------------------------------------- kernel 2
// grid = (128 i-tiles, jsplit j-slices); 8 waves per block.
// Gram matrix tile via V_WMMA_F32_16X16X4_F32 (4 chained, K=16),
// dist = v_sqrt_f32(ni + nj - 2G), then each lane (lane==f) accumulates
// sum_j sin(dist * freq_f) with __shfl broadcasts of the tile distances.
// B fragments for the next j-tile are prefetched behind the sin phase.
// Each j-slice writes an unscaled partial sum; the apply kernel reduces.
__global__ __launch_bounds__(256) void mrope_sin_node(
    const float* __restrict__ pos, const float* __restrict__ nrm,
    const float* __restrict__ log_freqs, float* __restrict__ part) {
  __shared__ float red[8][16][32];

  const int tid = threadIdx.x;
  const int wave = tid >> 5;
  const int lane = tid & 31;
  const int m = lane & 15;    // tile row for A frag / tile col for B frag
  const int khalf = lane >> 4;
  const int i_base = blockIdx.x * 16;

  // j-tile work partition across (blockIdx.y, wave)
  const int wseq = blockIdx.y * 8 + wave;
  const int wstride = gridDim.y * 8;

  // lane == frequency index
  const float freq = __expf(log_freqs[lane]);

  // i-row norms: acc VGPR r holds D-matrix row M = r + 8*khalf
  float ni[8];
#pragma unroll
  for (int r = 0; r < 8; ++r) ni[r] = nrm[i_base + r + 8 * khalf];

  float acc[16];
#pragma unroll
  for (int i = 0; i < 16; ++i) acc[i] = 0.f;

  // A fragment (fixed for whole block): A[M=m][K = 4*kc + r + 2*khalf]
  v2f afrag[4];
  {
    const float* pa = pos + (size_t)(i_base + m) * PP;
#pragma unroll
    for (int kc = 0; kc < 4; ++kc) {
      afrag[kc].x = pa[4 * kc + 2 * khalf + 0];
      afrag[kc].y = pa[4 * kc + 2 * khalf + 1];
    }
  }

  // prefetched B fragments / col norm for the current j-tile
  v2f bfrag[4];
  float njv;
  {
    const int j_base = wseq * 16;
    const float* pb = pos + (size_t)(j_base + m) * PP;
    njv = nrm[j_base + m];
#pragma unroll
    for (int kc = 0; kc < 4; ++kc) {
      bfrag[kc].x = pb[4 * kc + 2 * khalf + 0];
      bfrag[kc].y = pb[4 * kc + 2 * khalf + 1];
    }
  }

  for (int jt = wseq; jt < NN / 16; jt += wstride) {
    // Gram tile: 4 chained K=4 WMMAs, fragments already in registers
    v8f c = {0.f, 0.f, 0.f, 0.f, 0.f, 0.f, 0.f, 0.f};
#pragma unroll
    for (int kc = 0; kc < 4; ++kc) {
      c = __builtin_amdgcn_wmma_f32_16x16x4_f32(false, afrag[kc], false,
                                                bfrag[kc], (short)0, c, false,
                                                false);
    }

    // kick off next tile's loads; they retire under the sin phase
    const float nj_cur = njv;
    {
      const int jn = (jt + wstride < NN / 16) ? jt + wstride : wseq;  // clamp
      const int j_base = jn * 16;
      const float* pb = pos + (size_t)(j_base + m) * PP;
      njv = nrm[j_base + m];
#pragma unroll
      for (int kc = 0; kc < 4; ++kc) {
        bfrag[kc].x = pb[4 * kc + 2 * khalf + 0];
        bfrag[kc].y = pb[4 * kc + 2 * khalf + 1];
      }
    }

    // distances for the 8 (row, col) pairs this lane owns (fast v_sqrt_f32)
    float dist[8];
#pragma unroll
    for (int r = 0; r < 8; ++r) {
      float d2 = ni[r] + nj_cur - 2.0f * c[r];
      dist[r] = __builtin_amdgcn_sqrtf(fmaxf(d2, 1e-12f));
    }

    // every lane accumulates all 256 tile pairs at its own frequency.
    // j outer (rolled), i inner (unrolled): 16 independent shfl+sin chains
    // per step -> trans/VALU co-execution, bpermute latency hidden.
    for (int j = 0; j < 16; ++j) {
#pragma unroll
      for (int i = 0; i < 16; ++i) {
        const int src = ((i >> 3) << 4) + j;  // half-wave owning row i, col j
        float dv = __shfl(dist[i & 7], src, 32);
        acc[i] += __sinf(dv * freq);
      }
    }
  }

  // cross-wave reduction (within this j-slice)
#pragma unroll
  for (int i = 0; i < 16; ++i) red[wave][i][lane] = acc[i];
  __syncthreads();

  float* slice = part + (size_t)blockIdx.y * NN * FF;
#pragma unroll
  for (int p = 0; p < 2; ++p) {
    int idx = tid + p * 256;  // 16*32 = 512 outputs per block
    int i = idx >> 5;
    int f = idx & 31;
    float s = 0.f;
#pragma unroll
    for (int w = 0; w < 8; ++w) s += red[w][i][f];
    slice[(size_t)(i_base + i) * FF + f] = s;  // unscaled partial
  }
}

// ---------------------------------------------------------------- kernel 3
// q_rot / k_rot streaming apply; folds the jsplit partial reduction in.
// One thread per (n, h, f), f in [0,32).
__global__ __launch_bounds__(256) void mrope_apply(
    const float* __restrict__ q, const float* __restrict__ k,
    const float* __restrict__ part, const float* __restrict__ log_freqs,
    float* __restrict__ out, int jsplit) {
  int idx = blockIdx.x * 256 + threadIdx.x;  // NN*HH*32 total
  int f = idx & 31;
  int nh = idx >> 5;  // n*HH + h
  int n = nh >> 4;    // HH == 16

  float freq = __expf(log_freqs[f]);
  float cv = __cosf(1e-6f * freq);  // diag(dist) == sqrt(1e-12) == 1e-6

  float sv = 0.f;
  for (int s = 0; s < jsplit; ++s)
    sv += part[(size_t)s * NN * FF + (size_t)n * FF + f];
  sv *= (1.0f / NN);

  size_t base = (size_t)nh * DD;
  float q1 = q[base + f], q2 = q[base + 32 + f];
  float k1 = k[base + f], k2 = k[base + 32 + f];

  float* qo = out;
  float* ko = out + (size_t)NN * HH * DD;
  qo[base + f] = q1 * cv - q2 * sv;
  qo[base + 32 + f] = q1 * sv + q2 * cv;
  ko[base + f] = k1 * cv - k2 * sv;
  ko[base + 32 + f] = k1 * sv + k2 * cv;
}

// ---------------------------------------------------------------- launcher
extern "C" void kernel_launch(void* const* d_in, const int* in_sizes, int n_in,
                              void* d_out, int out_size, void* d_ws,
                              size_t ws_size, hipStream_t stream) {
  (void)in_sizes;
  (void)n_in;
  (void)out_size;

  const float* q = (const float*)d_in[0];
  const float* k = (const float*)d_in[1];
  const float* pos = (const float*)d_in[2];
  const float* logf = (const float*)d_in[3];
  float* out = (float*)d_out;

  // choose j-split (grid multiplier) from available workspace;
  // ws_size is fixed across calls -> deterministic.
  int jsplit = 1;
  if (ws_size >= (size_t)(NN + 4 * NN * FF) * sizeof(float)) jsplit = 4;
  else if (ws_size >= (size_t)(NN + 2 * NN * FF) * sizeof(float)) jsplit = 2;

  float* nrm = (float*)d_ws;   // NN floats
  float* part = nrm + NN;      // jsplit * NN * FF floats

  mrope_norms<<<NN / 256, 256, 0, stream>>>(pos, nrm);
  mrope_sin_node<<<dim3(NN / 16, jsplit), 256, 0, stream>>>(pos, nrm, logf,
                                                            part);
  mrope_apply<<<(NN * HH * 32) / 256, 256, 0, stream>>>(q, k, part, logf, out,
                                                        jsplit);
}